// comboLoss_6382321402277
// MI455X (gfx1250) — compile-verified
//
#include <hip/hip_runtime.h>

typedef __attribute__((ext_vector_type(2))) float v2f;
typedef __attribute__((ext_vector_type(8))) float v8f;

#define BATCH 256
#define SEQ   8192
// ALPHA = BETA = 1.0

// Wave32 cross-lane sum of TWO independent scalars using one
// V_WMMA_F32_16X16X4_F32.
// A (16x4 f32, 2 VGPRs/lane): VGPR0=A[lane%16][lane<16?0:2]=vx,
//                             VGPR1=A[lane%16][lane<16?1:3]=vy.
// B (4x16, 2 VGPRs): selector matrix B[k][n] = 1 iff (k even) == (n<8),
// so D[m][n<8] = A[m][0]+A[m][2] (x-channel), D[m][n>=8] = A[m][1]+A[m][3].
// Summing the 8 D VGPRs per lane gives half-column sums; one xor-16 add
// completes the column sum: lanes with (lane&15)<8 hold sum(vx over wave),
// lanes with (lane&15)>=8 hold sum(vy over wave).
__device__ inline void wave_sum2_wmma(float vx, float vy, int lane,
                                      float& sum_x, float& sum_y) {
    v2f A;
    A.x = vx;
    A.y = vy;
    const int n = lane & 15;
    v2f Bsel;
    Bsel.x = (n < 8) ? 1.0f : 0.0f;   // rows K=0 (lanes<16) / K=2 (lanes>=16)
    Bsel.y = (n < 8) ? 0.0f : 1.0f;   // rows K=1 / K=3
    v8f C = {};
    v8f D = __builtin_amdgcn_wmma_f32_16x16x4_f32(
        /*neg_a=*/false, A, /*neg_b=*/false, Bsel,
        /*c_mod=*/(short)0, C, /*reuse_a=*/false, /*reuse_b=*/false);
    float s = D[0] + D[1] + D[2] + D[3] + D[4] + D[5] + D[6] + D[7];
    s += __shfl_xor(s, 16);
    sum_x = __shfl(s, 0);   // a column with n<8
    sum_y = __shfl(s, 8);   // a column with n>=8
}

__global__ __launch_bounds__(256) void combo_row_kernel(
    const float* __restrict__ start_logits,
    const float* __restrict__ end_logits,
    const int* __restrict__ start_positions,
    const int* __restrict__ end_positions,
    const float* __restrict__ pt,
    const float* __restrict__ tagging,
    float* __restrict__ partials) {
    const int b    = blockIdx.x;
    const int tid  = threadIdx.x;   // 0..255
    const int lane = tid & 31;
    const int wave = tid >> 5;      // 0..7

    const float* __restrict__ xs = start_logits + (size_t)b * SEQ;
    const float* __restrict__ xe = end_logits   + (size_t)b * SEQ;
    const float* __restrict__ pp = pt           + (size_t)b * SEQ;
    const float* __restrict__ tg = tagging      + (size_t)b * SEQ;

    __shared__ float red_a[8];
    __shared__ float red_b[8];
    __shared__ float red_c[8];

    // ---------- Pass 1: stream all four rows once (b128 loads) ----------
    // Thread covers 8 strided float4s per tensor: 256 thr * 4 * 8 = 8192.
    float mx_s = -__builtin_inff();
    float mx_e = -__builtin_inff();
    float bce  = 0.0f;   // accumulates +[T*log(p) + (1-T)*log(1-p)]
#pragma unroll
    for (int i = 0; i < 8; ++i) {
        const int idx = (tid + (i << 8)) << 2;   // float4-aligned
        float4 a = *(const float4*)(xs + idx);
        float4 e = *(const float4*)(xe + idx);
        float4 p = *(const float4*)(pp + idx);
        float4 t = *(const float4*)(tg + idx);
        mx_s = fmaxf(mx_s, fmaxf(fmaxf(a.x, a.y), fmaxf(a.z, a.w)));
        mx_e = fmaxf(mx_e, fmaxf(fmaxf(e.x, e.y), fmaxf(e.z, e.w)));
        bce += t.x * __logf(p.x) + (1.0f - t.x) * __logf(1.0f - p.x);
        bce += t.y * __logf(p.y) + (1.0f - t.y) * __logf(1.0f - p.y);
        bce += t.z * __logf(p.z) + (1.0f - t.z) * __logf(1.0f - p.z);
        bce += t.w * __logf(p.w) + (1.0f - t.w) * __logf(1.0f - p.w);
    }

    // Block-wide max of (mx_s, mx_e): wave shuffle tree then LDS combine.
#pragma unroll
    for (int off = 16; off > 0; off >>= 1) {
        mx_s = fmaxf(mx_s, __shfl_xor(mx_s, off));
        mx_e = fmaxf(mx_e, __shfl_xor(mx_e, off));
    }
    if (lane == 0) { red_a[wave] = mx_s; red_b[wave] = mx_e; }
    __syncthreads();
    float Ms = red_a[0], Me = red_b[0];
#pragma unroll
    for (int w = 1; w < 8; ++w) {
        Ms = fmaxf(Ms, red_a[w]);
        Me = fmaxf(Me, red_b[w]);
    }
    __syncthreads();

    // ---------- Pass 2: sum exp(x - M); rows are hot in L1/L2 ----------
    float es = 0.0f, ee = 0.0f;
#pragma unroll
    for (int i = 0; i < 8; ++i) {
        const int idx = (tid + (i << 8)) << 2;
        float4 a = *(const float4*)(xs + idx);
        float4 e = *(const float4*)(xe + idx);
        es += __expf(a.x - Ms) + __expf(a.y - Ms) +
              __expf(a.z - Ms) + __expf(a.w - Ms);
        ee += __expf(e.x - Me) + __expf(e.y - Me) +
              __expf(e.z - Me) + __expf(e.w - Me);
    }

    // Wave-level sums via WMMA: (es, ee) in one op, (bce, 0) in another.
    float ws_es, ws_ee, ws_bce, ws_dummy;
    wave_sum2_wmma(es, ee, lane, ws_es, ws_ee);
    wave_sum2_wmma(bce, 0.0f, lane, ws_bce, ws_dummy);
    (void)ws_dummy;

    if (lane == 0) { red_a[wave] = ws_es; red_b[wave] = ws_ee; red_c[wave] = ws_bce; }
    __syncthreads();

    if (tid == 0) {
        float SS = 0.0f, SE = 0.0f, BC = 0.0f;
#pragma unroll
        for (int w = 0; w < 8; ++w) { SS += red_a[w]; SE += red_b[w]; BC += red_c[w]; }
        const float xs_gold = xs[start_positions[b]];
        const float xe_gold = xe[end_positions[b]];
        const float lp_start = xs_gold - (Ms + __logf(SS));
        const float lp_end   = xe_gold - (Me + __logf(SE));
        // partial_b = -(lp_s+lp_e)/(2B)  +  (-BC)/(B*S)
        const float inv2B = 1.0f / (2.0f * (float)BATCH);
        const float invBS = 1.0f / ((float)BATCH * (float)SEQ);
        partials[b] = -(lp_start + lp_end) * inv2B - BC * invBS;
    }
}

__global__ __launch_bounds__(32) void combo_final_kernel(
    const float* __restrict__ partials, float* __restrict__ out) {
    const int lane = threadIdx.x;
    float s = 0.0f;
#pragma unroll
    for (int i = 0; i < 8; ++i) s += partials[lane + (i << 5)];
    float sx, sy;
    wave_sum2_wmma(s, 0.0f, lane, sx, sy);
    (void)sy;
    if (lane == 0) out[0] = sx;
}

extern "C" void kernel_launch(void* const* d_in, const int* in_sizes, int n_in,
                              void* d_out, int out_size, void* d_ws, size_t ws_size,
                              hipStream_t stream) {
    (void)in_sizes; (void)n_in; (void)out_size; (void)ws_size;
    const float* start_logits    = (const float*)d_in[0];
    const float* end_logits      = (const float*)d_in[1];
    const int*   start_positions = (const int*)d_in[2];
    const int*   end_positions   = (const int*)d_in[3];
    const float* pt              = (const float*)d_in[4];
    const float* tagging         = (const float*)d_in[5];
    float* partials = (float*)d_ws;   // 256 floats

    combo_row_kernel<<<BATCH, 256, 0, stream>>>(
        start_logits, end_logits, start_positions, end_positions,
        pt, tagging, partials);
    combo_final_kernel<<<1, 32, 0, stream>>>(partials, (float*)d_out);
}